// Quantizer_59244778881303
// MI455X (gfx1250) — compile-verified
//
#include <hip/hip_runtime.h>
#include <math.h>

typedef __attribute__((ext_vector_type(2))) float v2f;
typedef __attribute__((ext_vector_type(8))) float v8f;

#define N_E    4096
#define E_DIM  64
#define N_TOK  16384
#define BETA   0.25f

// -------------------------------------------------------------------------
// Kernel 1: codebook norms + zero accumulators (ws is not re-poisoned
// between replays, so we must re-init every launch).
// -------------------------------------------------------------------------
__global__ __launch_bounds__(256) void vq_norms_init(const float* __restrict__ cb,
                                                     float* __restrict__ norms,
                                                     float* __restrict__ counts,
                                                     float* __restrict__ lossAcc) {
  int k = blockIdx.x * blockDim.x + threadIdx.x;
  if (k < N_E) {
    const float4* row = (const float4*)(cb + (size_t)k * E_DIM);
    float s = 0.0f;
#pragma unroll
    for (int i = 0; i < E_DIM / 4; ++i) {
      float4 v = row[i];
      s += v.x * v.x + v.y * v.y + v.z * v.z + v.w * v.w;
    }
    norms[k]  = s;
    counts[k] = 0.0f;
  }
  if (k == 0) *lossAcc = 0.0f;
}

// -------------------------------------------------------------------------
// Kernel 2: fused (-2 z.e^T + ||e||^2) GEMM + row argmin via WMMA f32.
// One wave32 owns a 16-token tile; loops over all 4096 codes in 16-code
// tiles. A/B fragments follow the ISA 16x4 f32 layout: lanes 0-15 carry
// K=0,1 (2 VGPRs), lanes 16-31 carry K=2,3.
// -------------------------------------------------------------------------
__global__ __launch_bounds__(256) void vq_main(const float* __restrict__ z,
                                               const float* __restrict__ cb,
                                               const float* __restrict__ norms,
                                               int*   __restrict__ idx_out,
                                               float* __restrict__ idxf_out,
                                               float* __restrict__ counts) {
  const int lane     = threadIdx.x & 31;      // wave32
  const int wave     = threadIdx.x >> 5;
  const int tileBase = (blockIdx.x * 8 + wave) * 16;   // 16 tokens per wave
  const int nl       = lane & 15;             // N / M position within tile
  const int half     = lane >> 4;             // which K-pair this lane holds
  const int koff     = half * 2;

  // A fragment: token tile, 16 K-steps of 16x4 (float2 per lane per step)
  v2f a[16];
  const float* zrow = z + (size_t)(tileBase + nl) * E_DIM + koff;
#pragma unroll
  for (int k = 0; k < 16; ++k)
    a[k] = *(const v2f*)(zrow + 4 * k);

  float bestVal[8];
  int   bestIdx[8];
#pragma unroll
  for (int r = 0; r < 8; ++r) { bestVal[r] = 3.0e38f; bestIdx[r] = 0; }

  for (int c0 = 0; c0 < N_E; c0 += 16) {
    if (c0 + 16 < N_E)  // pull next codebook tile toward the WGP
      __builtin_prefetch(cb + (size_t)(c0 + 16 + nl) * E_DIM, 0, 3);

    const float* brow = cb + (size_t)(c0 + nl) * E_DIM + koff;
    v8f acc = {};
#pragma unroll
    for (int k = 0; k < 16; ++k) {
      v2f b = *(const v2f*)(brow + 4 * k);
      // D = A(16x4,f32) x B(4x16,f32) + C  -> v_wmma_f32_16x16x4_f32
      acc = __builtin_amdgcn_wmma_f32_16x16x4_f32(
          /*neg_a=*/false, a[k], /*neg_b=*/false, b,
          /*c_mod=*/(short)0, acc, /*reuse_a=*/false, /*reuse_b=*/false);
    }

    // acc[r] = dot(z[m], e[n]) with m = r + 8*half, n = c0 + nl.
    // distance (modulo the per-row constant ||z||^2): ||e||^2 - 2*dot
    float nrm = norms[c0 + nl];
#pragma unroll
    for (int r = 0; r < 8; ++r) {
      float d = __builtin_fmaf(-2.0f, acc[r], nrm);
      if (d < bestVal[r]) { bestVal[r] = d; bestIdx[r] = c0 + nl; }
    }
  }

  // min-reduce across the 16 lanes of each half (n varies, m fixed per reg);
  // tie-break to the smallest index to match argmin semantics.
#pragma unroll
  for (int off = 1; off < 16; off <<= 1) {
#pragma unroll
    for (int r = 0; r < 8; ++r) {
      float ov = __shfl_xor(bestVal[r], off, 32);
      int   oi = __shfl_xor(bestIdx[r], off, 32);
      if (ov < bestVal[r] || (ov == bestVal[r] && oi < bestIdx[r])) {
        bestVal[r] = ov; bestIdx[r] = oi;
      }
    }
  }

  if (nl == 0) {
    int mbase = tileBase + (half << 3);
#pragma unroll
    for (int r = 0; r < 8; ++r) {
      idx_out[mbase + r]  = bestIdx[r];
      idxf_out[mbase + r] = (float)bestIdx[r];
      atomicAdd(&counts[bestIdx[r]], 1.0f);
    }
  }
}

// -------------------------------------------------------------------------
// Kernel 3: gather z_q = codebook[idx], write output, accumulate sum of
// squared error for the loss.
// -------------------------------------------------------------------------
__global__ __launch_bounds__(256) void vq_gather(const float* __restrict__ z,
                                                 const float* __restrict__ cb,
                                                 const int* __restrict__ idx,
                                                 float* __restrict__ zq_out,
                                                 float* __restrict__ lossAcc) {
  int gid = blockIdx.x * 256 + threadIdx.x;   // N_TOK*E_DIM/2 threads
  int tok = gid >> 5;                         // 32 threads per token
  int off = (gid & 31) * 2;                   // float2 per thread
  int code = idx[tok];

  v2f e  = *(const v2f*)(cb + (size_t)code * E_DIM + off);
  v2f zv = *(const v2f*)(z  + (size_t)tok  * E_DIM + off);
  *(v2f*)(zq_out + (size_t)tok * E_DIM + off) = e;

  float dx = e.x - zv.x, dy = e.y - zv.y;
  float s = dx * dx + dy * dy;

  __shared__ float red[256];
  red[threadIdx.x] = s;
  __syncthreads();
  for (int st = 128; st > 0; st >>= 1) {
    if (threadIdx.x < st) red[threadIdx.x] += red[threadIdx.x + st];
    __syncthreads();
  }
  if (threadIdx.x == 0) atomicAdd(lossAcc, red[0]);
}

// -------------------------------------------------------------------------
// Kernel 4: perplexity + final loss scalar.
// -------------------------------------------------------------------------
__global__ __launch_bounds__(256) void vq_finalize(const float* __restrict__ counts,
                                                   const float* __restrict__ lossAcc,
                                                   float* __restrict__ loss_out,
                                                   float* __restrict__ ppl_out) {
  __shared__ float red[256];
  float s = 0.0f;
  for (int k = threadIdx.x; k < N_E; k += 256) {
    float e = counts[k] * (1.0f / (float)N_TOK);
    s += e * logf(e + 1e-10f);
  }
  red[threadIdx.x] = s;
  __syncthreads();
  for (int st = 128; st > 0; st >>= 1) {
    if (threadIdx.x < st) red[threadIdx.x] += red[threadIdx.x + st];
    __syncthreads();
  }
  if (threadIdx.x == 0) {
    *ppl_out  = expf(-red[0]);
    // loss = mean((zq-z)^2) + BETA*mean((zq-z)^2)   (sg is identity in fwd)
    *loss_out = (*lossAcc) * (1.0f + BETA) * (1.0f / (float)(N_TOK * E_DIM));
  }
}

// -------------------------------------------------------------------------
// d_out layout (floats, return-order concat):
//   [0]                      loss
//   [1 .. 1+N_TOK*E_DIM)     z_q_st  (== z_q forward value)
//   [.. +N_TOK)              idx (as float)
//   [last]                   perplexity
// ws layout (floats): norms[4096] | idx[16384 int] | counts[4096] | lossAcc[1]
// -------------------------------------------------------------------------
extern "C" void kernel_launch(void* const* d_in, const int* in_sizes, int n_in,
                              void* d_out, int out_size, void* d_ws, size_t ws_size,
                              hipStream_t stream) {
  const float* z  = (const float*)d_in[0];
  const float* cb = (const float*)d_in[1];

  float* out      = (float*)d_out;
  float* loss_out = out;
  float* zq_out   = out + 1;
  float* idxf_out = out + 1 + N_TOK * E_DIM;
  float* ppl_out  = out + 1 + N_TOK * E_DIM + N_TOK;

  float* ws     = (float*)d_ws;
  float* norms  = ws;
  int*   idx    = (int*)(ws + N_E);
  float* counts = ws + N_E + N_TOK;
  float* lossA  = ws + N_E + N_TOK + N_E;

  vq_norms_init<<<N_E / 256, 256, 0, stream>>>(cb, norms, counts, lossA);
  vq_main<<<N_TOK / (16 * 8), 256, 0, stream>>>(z, cb, norms, idx, idxf_out, counts);
  vq_gather<<<(N_TOK * E_DIM / 2) / 256, 256, 0, stream>>>(z, cb, idx, zq_out, lossA);
  vq_finalize<<<1, 256, 0, stream>>>(counts, lossA, loss_out, ppl_out);
}